// DFS_Mixer_87497073754187
// MI455X (gfx1250) — compile-verified
//
#include <hip/hip_runtime.h>

typedef float v2f __attribute__((ext_vector_type(2)));
typedef float v8f __attribute__((ext_vector_type(8)));
typedef float f4  __attribute__((ext_vector_type(4)));

#define B_  8
#define K_  8
#define C_  256
#define HW_ 4096

// ---------------------------------------------------------------------------
// Pass 1: one block per (b,c). Wave w computes dot_k / ||CR_k||^2 for k = w
// over the 4096-long HW axis with float4 streaming loads, wave32-shuffle
// reduces, then thread 0 runs the 8-wide softmax and writes weights[b,k,c].
// ---------------------------------------------------------------------------
__global__ __launch_bounds__(256) void sim_softmax_kernel(
    const float* __restrict__ CR, const float* __restrict__ CT,
    float* __restrict__ weights)
{
    __shared__ float sdot[K_];
    __shared__ float snrm[K_];

    const int bc   = blockIdx.x;
    const int b    = bc / C_;
    const int c    = bc % C_;
    const int k    = threadIdx.x >> 5;   // wave id == style index
    const int lane = threadIdx.x & 31;

    const f4* ct = (const f4*)(CT + ((size_t)b * C_ + c) * (size_t)HW_);
    const f4* cr = (const f4*)(CR + (((size_t)b * K_ + k) * C_ + c) * (size_t)HW_);

    float dot = 0.f, nrm = 0.f;
    #pragma unroll 4
    for (int i = lane; i < HW_ / 4; i += 32) {
        f4 r = __builtin_nontemporal_load(&cr[i]);   // streamed once: NT
        f4 t = ct[i];                                // reused 8x: keep cached
        dot += r.x * t.x + r.y * t.y + r.z * t.z + r.w * t.w;
        nrm += r.x * r.x + r.y * r.y + r.z * r.z + r.w * r.w;
    }
    #pragma unroll
    for (int off = 16; off >= 1; off >>= 1) {
        dot += __shfl_xor(dot, off, 32);
        nrm += __shfl_xor(nrm, off, 32);
    }
    if (lane == 0) { sdot[k] = dot; snrm[k] = nrm; }
    __syncthreads();

    if (threadIdx.x == 0) {
        float sim[K_], mx = -1e30f;
        #pragma unroll
        for (int kk = 0; kk < K_; ++kk) {
            sim[kk] = 2.0f * sdot[kk] * rsqrtf(snrm[kk]);
            mx = fmaxf(mx, sim[kk]);
        }
        float e[K_], se = 0.f;
        #pragma unroll
        for (int kk = 0; kk < K_; ++kk) { e[kk] = __expf(sim[kk] - mx); se += e[kk]; }
        const float inv = 1.0f / se;
        #pragma unroll
        for (int kk = 0; kk < K_; ++kk)
            weights[((size_t)b * K_ + kk) * C_ + c] = e[kk] * inv;
    }
}

// ---------------------------------------------------------------------------
// Pass 2: one block per (b,c). out[hw] = sum_k w[k] * IR[k,hw] mapped onto
// V_WMMA_F32_16X16X4_F32:  A(16x4) = IR[hw_tile(16) x k(4)],
// B(4x16)[k,n] = w[k] (broadcast columns), two chained WMMAs cover k=0..7.
// A layout (ISA 7.12.2, 32-bit A 16x4): lanes 0-15 -> {K0,K1}, 16-31 -> {K2,K3}.
// D layout: lane 0 holds out[hw0..hw0+7] in v0..7, lane 16 holds out[hw0+8..15].
// ---------------------------------------------------------------------------
__global__ __launch_bounds__(256) void mix_wmma_kernel(
    const float* __restrict__ IR, const float* __restrict__ weights,
    float* __restrict__ out)
{
    const int bc   = blockIdx.x;
    const int b    = bc / C_;
    const int c    = bc % C_;
    const int wave = threadIdx.x >> 5;
    const int lane = threadIdx.x & 31;
    const int m    = lane & 15;            // row of A (hw within tile)
    const int kb   = (lane >> 4) << 1;     // 0 (lanes 0-15) or 2 (lanes 16-31)

    const size_t CHW = (size_t)C_ * HW_;
    const float* ir = IR + ((size_t)b * K_ * C_ + c) * (size_t)HW_;  // k = 0 plane
    float*       o  = out + ((size_t)b * C_ + c) * (size_t)HW_;
    const float* wp = weights + (size_t)b * K_ * C_ + c;             // w[k] = wp[k*C]

    v2f b_lo, b_hi;                        // B operand: rows K=kb, kb+1 (and +4)
    b_lo.x = wp[(size_t)(kb + 0) * C_];
    b_lo.y = wp[(size_t)(kb + 1) * C_];
    b_hi.x = wp[(size_t)(kb + 4) * C_];
    b_hi.y = wp[(size_t)(kb + 5) * C_];

    for (int tile = wave; tile < HW_ / 16; tile += 8) {
        const int hw0 = tile * 16;
        const float* pa = ir + hw0 + m;

        v2f a_lo, a_hi;                    // each half-wave load = 64B coalesced
        a_lo.x = __builtin_nontemporal_load(pa + (size_t)(kb + 0) * CHW);
        a_lo.y = __builtin_nontemporal_load(pa + (size_t)(kb + 1) * CHW);
        a_hi.x = __builtin_nontemporal_load(pa + (size_t)(kb + 4) * CHW);
        a_hi.y = __builtin_nontemporal_load(pa + (size_t)(kb + 5) * CHW);

        v8f acc = {0.f, 0.f, 0.f, 0.f, 0.f, 0.f, 0.f, 0.f};
        acc = __builtin_amdgcn_wmma_f32_16x16x4_f32(
                  false, a_lo, false, b_lo, (short)0, acc, false, false);
        acc = __builtin_amdgcn_wmma_f32_16x16x4_f32(
                  false, a_hi, false, b_hi, (short)0, acc, false, false);

        if (m == 0) {                      // lanes 0 and 16 own the full column
            float* p = o + hw0 + ((lane >> 4) << 3);
            f4 lo = {acc[0], acc[1], acc[2], acc[3]};
            f4 hi = {acc[4], acc[5], acc[6], acc[7]};
            __builtin_nontemporal_store(lo, (f4*)p);
            __builtin_nontemporal_store(hi, (f4*)(p + 4));
        }
    }
}

// ---------------------------------------------------------------------------
extern "C" void kernel_launch(void* const* d_in, const int* in_sizes, int n_in,
                              void* d_out, int out_size, void* d_ws, size_t ws_size,
                              hipStream_t stream) {
    (void)in_sizes; (void)n_in; (void)out_size; (void)ws_size;
    const float* IR = (const float*)d_in[0];   // [B,K,C,H,W]
    const float* CR = (const float*)d_in[1];   // [B,K,C,H,W]
    const float* CT = (const float*)d_in[2];   // [B,C,H,W]
    float* wts = (float*)d_ws;                 // [B,K,C] = 64 KB scratch
    float* outp = (float*)d_out;               // [B,C,H,W]

    sim_softmax_kernel<<<B_ * C_, 256, 0, stream>>>(CR, CT, wts);
    mix_wmma_kernel<<<B_ * C_, 256, 0, stream>>>(IR, wts, outp);
}